// UVMDBlock_56736517980845
// MI455X (gfx1250) — compile-verified
//
#include <hip/hip_runtime.h>
#include <hip/hip_bf16.h>

// ---------------------------------------------------------------------------
// Fused VMD block for MI455X (gfx1250, wave32).
// One workgroup (512 threads = 16 waves) owns one (b,c) row of length 4096.
//   1) gather row -> LDS; TDM stages DFT-64 + twiddle tables into LDS (async)
//   2) forward 4096-FFT as 64x64 factorization: two 64x64 matmul stages
//      (V_WMMA_F32_16X16X4_F32) + twiddle
//   3) 8*4 Gauss-Seidel ADMM steps, fully in registers (8 bins/thread);
//      TDM concurrently swaps inverse tables into LDS under the ADMM
//   4) per mode k: inverse 64x64-factored FFT (real part only) -> scatter out
// Tables (128 KB) built once per launch into d_ws; L2-resident, TDM'd to LDS.
// ---------------------------------------------------------------------------

typedef float v2f __attribute__((ext_vector_type(2)));
typedef float v8f __attribute__((ext_vector_type(8)));
typedef unsigned int v4u __attribute__((ext_vector_type(4)));
typedef int v4i __attribute__((ext_vector_type(4)));
typedef int v8i __attribute__((ext_vector_type(8)));

#define T_LEN   4096
#define KM      4
#define LL      8
#define NTHR    512           // 16 waves * 32 lanes
#define CCH     64            // channels
#define PI_F    3.14159265358979323846f

// ---- table offsets in d_ws (floats) ----
#define OFF_W64RE  0          // e^{-2pi i ab/64}  (forward DFT-64)
#define OFF_W64IM  4096
#define OFF_TWRE   8192       // e^{-2pi i ab/4096} (forward twiddle)
#define OFF_TWIM   12288
#define OFF_WCRE   16384      // e^{+2pi i ab/64} / 64 (inverse DFT-64, scaled)
#define OFF_WCIM   20480
#define OFF_ITWRE  24576      // e^{+2pi i ab/4096} (inverse twiddle)
#define OFF_ITWIM  28672

__device__ __forceinline__ v8f vzero8() {
  v8f z = {0.f, 0.f, 0.f, 0.f, 0.f, 0.f, 0.f, 0.f};
  return z;
}

// one 16x16x4 fp32 WMMA accumulate
__device__ __forceinline__ v8f wmma4(v8f acc, float a0, float a1, float b0, float b1) {
  v2f a; a.x = a0; a.y = a1;
  v2f b; b.x = b0; b.y = b1;
  return __builtin_amdgcn_wmma_f32_16x16x4_f32(false, a, false, b, (short)0, acc,
                                               false, false);
}

// TDM: async-copy 4096 contiguous f32 (16 KB) from global into LDS.
// D# per CDNA5 ISA: 1-row tile, data_size=4B,
// tile_dim0 = tensor_dim0 = 4096, type=2 ("image"). Tracked by TENSORcnt.
__device__ __forceinline__ void tdm_load_16kb(void* lds_dst, const float* src) {
  const unsigned long long ga = (unsigned long long)(uintptr_t)src;
  const unsigned int lds_off = (unsigned int)(uintptr_t)lds_dst; // addr[31:0] = LDS offset
  v4u g0;
  g0[0] = 1u;                                              // count=1, user D#
  g0[1] = lds_off;                                         // lds_addr (bytes)
  g0[2] = (unsigned int)ga;                                // global_addr[31:0]
  g0[3] = (unsigned int)((ga >> 32) & 0x1FFFFFFu) | (2u << 30); // addr[56:32] | type=2
  v8i g1;
  g1[0] = (int)(2u << 16);          // workgroup_mask=0, data_size=2 (4B)
  g1[1] = (int)(4096u << 16);       // tensor_dim0[15:0]=4096 at bits 63:48
  g1[2] = (int)(1u << 16);          // tensor_dim0 hi=0; tensor_dim1=1
  g1[3] = (int)(4096u << 16);       // tensor_dim1 hi=0; tile_dim0=4096
  g1[4] = 1;                        // tile_dim1=1, tile_dim2=0
  g1[5] = 4096;                     // tensor_dim0_stride = 4096 elements
  g1[6] = 0;                        // stride hi / dim1_stride lo
  g1[7] = 0;                        // dim1_stride hi
  v4i z4 = {0, 0, 0, 0};
#if defined(__clang_major__) && (__clang_major__ >= 23)
  v8i z8 = {0, 0, 0, 0, 0, 0, 0, 0};
  __builtin_amdgcn_tensor_load_to_lds(g0, g1, z4, z4, z8, 0);
#else
  __builtin_amdgcn_tensor_load_to_lds(g0, g1, z4, z4, 0);
#endif
}

// acc += sgn * op(A) * B over the 64x64x64 tile (mi,ni) owned by this wave.
// Fragment layouts per CDNA5 ISA 7.12.2 (fp32 16x16x4):
//   A: lane -> M = lane%16, vgpr r -> K = 2*(lane/16) + r
//   B: lane -> N = lane%16, vgpr r -> K = 2*(lane/16) + r
//   C: lane -> N = lane%16, vgpr r -> M = r + 8*(lane/16)
__device__ __forceinline__ v8f mm64(v8f acc, const float* __restrict__ A,
                                    const float* __restrict__ B,
                                    bool transA, float sgn,
                                    int mi, int ni, int lane) {
  const int mr = mi * 16 + (lane & 15);
  const int nr = ni * 16 + (lane & 15);
  const int kh = (lane >> 4) << 1;
#pragma unroll
  for (int kk = 0; kk < 16; ++kk) {
    const int k0 = (kk << 2) + kh;
    float a0, a1;
    if (transA) { a0 = A[k0 * 64 + mr]; a1 = A[(k0 + 1) * 64 + mr]; }
    else        { a0 = A[mr * 64 + k0]; a1 = A[mr * 64 + k0 + 1]; }
    const float b0 = B[k0 * 64 + nr];
    const float b1 = B[(k0 + 1) * 64 + nr];
    acc = wmma4(acc, sgn * a0, sgn * a1, b0, b1);
  }
  return acc;
}

__device__ __forceinline__ void st64(float* __restrict__ C, v8f acc,
                                     int mi, int ni, int lane) {
  const int nr = ni * 16 + (lane & 15);
  const int mb = mi * 16 + ((lane >> 4) << 3);
#pragma unroll
  for (int r = 0; r < 8; ++r) C[(mb + r) * 64 + nr] = acc[r];
}

// ---------------------------------------------------------------------------
// Table builder: 4096 (a,b) pairs, 8 tables.
// ---------------------------------------------------------------------------
__global__ void vmd_tables_kernel(float* __restrict__ tabs) {
  const int i = blockIdx.x * blockDim.x + threadIdx.x;
  if (i >= 4096) return;
  const int a = i >> 6, b = i & 63;
  const int m64 = (a * b) & 63;          // phase period 64
  const int mT  = a * b;                 // < 4096
  const float th64 = (2.0f * PI_F / 64.0f)   * (float)m64;
  const float thT  = (2.0f * PI_F / 4096.0f) * (float)mT;
  float c64 = cosf(th64), s64 = sinf(th64);
  float cT  = cosf(thT),  sT  = sinf(thT);
  tabs[OFF_W64RE + i] = c64;
  tabs[OFF_W64IM + i] = -s64;
  tabs[OFF_TWRE  + i] = cT;
  tabs[OFF_TWIM  + i] = -sT;
  tabs[OFF_WCRE  + i] = c64 * (1.0f / 64.0f);
  tabs[OFF_WCIM  + i] = s64 * (1.0f / 64.0f);
  tabs[OFF_ITWRE + i] = cT;
  tabs[OFF_ITWIM + i] = sT;
}

// ---------------------------------------------------------------------------
// Main fused kernel: one row per workgroup.
// ---------------------------------------------------------------------------
__global__ __launch_bounds__(NTHR)
void vmd_row_kernel(const float* __restrict__ x,     // [B, T, C]
                    const float* __restrict__ alpha, // [L, K]
                    const float* __restrict__ tau,   // [L]
                    const float* __restrict__ omega, // [K]
                    const float* __restrict__ tabs,  // tables in d_ws
                    float* __restrict__ out)         // [B, K, T, C]
{
  __shared__ float sRe[T_LEN];   // spectrum / time buffer          16 KB
  __shared__ float sIm[T_LEN];   //                                 16 KB
  __shared__ float tRe[T_LEN];   // matmul-stage scratch            16 KB
  __shared__ float tIm[T_LEN];   //                                 16 KB
  __shared__ float dftR[T_LEN];  // DFT-64 table (fwd then inv)     16 KB
  __shared__ float dftI[T_LEN];  //                                 16 KB
  __shared__ float twR[T_LEN];   // twiddle table (fwd then inv)    16 KB
  __shared__ float twI[T_LEN];   //                                 16 KB

  const int tid  = threadIdx.x;
  const int lane = tid & 31;
  const int wave = tid >> 5;     // 0..15
  const int mi   = wave >> 2;
  const int ni   = wave & 3;

  const int row = blockIdx.x;    // 0..2047
  const int b   = row >> 6;
  const int c   = row & 63;

  // ---- 0) kick off async TDM of forward tables into LDS (wave 0 only) ----
  if (wave == 0) {
    tdm_load_16kb(dftR, tabs + OFF_W64RE);
    tdm_load_16kb(dftI, tabs + OFF_W64IM);
    tdm_load_16kb(twR,  tabs + OFF_TWRE);
    tdm_load_16kb(twI,  tabs + OFF_TWIM);
  }

  // ---- 1) gather row into LDS (x[b, t, c], stride C) under the TDM ----
  const float* xrow = x + ((size_t)b * T_LEN) * CCH + c;
  __builtin_prefetch(xrow + (size_t)tid * 8 * CCH, 0, 0);
  for (int i = tid; i < T_LEN; i += NTHR) sRe[i] = xrow[(size_t)i * CCH];

  if (wave == 0) __builtin_amdgcn_s_wait_tensorcnt(0);
  __syncthreads();

  // ---- 2) forward FFT: t = t0 + 64*t1, f = f0 + 64*f1 ----
  // stage 1: G[t0][f0] = sum_t1 M[t1][t0] * W64[t1][f0]  (M real) -> scratch
  {
    v8f gr = vzero8(), gi = vzero8();
    gr = mm64(gr, sRe, dftR, true,  1.f, mi, ni, lane);
    gi = mm64(gi, sRe, dftI, true,  1.f, mi, ni, lane);
    st64(tRe, gr, mi, ni, lane);
    st64(tIm, gi, mi, ni, lane);
  }
  __syncthreads();
  // twiddle: H = G .* e^{-2pi i t0 f0 / 4096}
#pragma unroll
  for (int j = 0; j < 8; ++j) {
    const int i = tid * 8 + j;
    const float gr = tRe[i], gi = tIm[i];
    const float wr = twR[i], wi = twI[i];
    tRe[i] = gr * wr - gi * wi;
    tIm[i] = gr * wi + gi * wr;
  }
  __syncthreads();
  // stage 2: X[f1][f0] = sum_t0 W64[t0][f1] * H[t0][f0]  (W64 symmetric)
  {
    v8f xr = vzero8(), xi = vzero8();
    xr = mm64(xr, dftR, tRe, false,  1.f, mi, ni, lane);
    xr = mm64(xr, dftI, tIm, false, -1.f, mi, ni, lane);
    xi = mm64(xi, dftR, tIm, false,  1.f, mi, ni, lane);
    xi = mm64(xi, dftI, tRe, false,  1.f, mi, ni, lane);
    st64(sRe, xr, mi, ni, lane);
    st64(sIm, xi, mi, ni, lane);
  }
  __syncthreads();

  // ---- swap in the inverse tables, hidden under the ADMM phase ----
  if (wave == 0) {
    tdm_load_16kb(dftR, tabs + OFF_WCRE);
    tdm_load_16kb(dftI, tabs + OFF_WCIM);
    tdm_load_16kb(twR,  tabs + OFF_ITWRE);
    tdm_load_16kb(twI,  tabs + OFF_ITWIM);
  }

  // ---- 3) ADMM on full spectrum, 8 bins/thread, all in registers ----
  float xr_[8], xi_[8], lr_[8], li_[8], ur_[KM][8], ui_[KM][8], frq[8];
#pragma unroll
  for (int j = 0; j < 8; ++j) {
    const int f = tid * 8 + j;
    xr_[j] = sRe[f];
    xi_[j] = sIm[f];
    lr_[j] = 0.f; li_[j] = 0.f;
    const int fm = (f <= 2048) ? f : (4096 - f);   // mirrored (Hermitian) freq
    frq[j] = (float)fm * (1.0f / 4096.0f);
#pragma unroll
    for (int k = 0; k < KM; ++k) { ur_[k][j] = 0.f; ui_[k][j] = 0.f; }
  }
  float om[KM];
#pragma unroll
  for (int k = 0; k < KM; ++k) om[k] = omega[k];

  for (int l = 0; l < LL; ++l) {
#pragma unroll
    for (int k = 0; k < KM; ++k) {
      const float al = alpha[l * KM + k];
#pragma unroll
      for (int j = 0; j < 8; ++j) {
        const float osr = ur_[0][j] + ur_[1][j] + ur_[2][j] + ur_[3][j] - ur_[k][j];
        const float osi = ui_[0][j] + ui_[1][j] + ui_[2][j] + ui_[3][j] - ui_[k][j];
        const float rr  = xr_[j] - osr + 0.5f * lr_[j];
        const float ri  = xi_[j] - osi + 0.5f * li_[j];
        const float d   = frq[j] - om[k];
        const float inv = 1.0f / (1.0f + al * d * d);
        ur_[k][j] = rr * inv;
        ui_[k][j] = ri * inv;
      }
    }
    const float tl = tau[l];
#pragma unroll
    for (int j = 0; j < 8; ++j) {
      const float sr = ur_[0][j] + ur_[1][j] + ur_[2][j] + ur_[3][j];
      const float si = ui_[0][j] + ui_[1][j] + ui_[2][j] + ui_[3][j];
      lr_[j] += tl * (xr_[j] - sr);
      li_[j] += tl * (xi_[j] - si);
    }
  }

  // inverse tables must have landed before the first inverse matmul
  if (wave == 0) __builtin_amdgcn_s_wait_tensorcnt(0);
  __syncthreads();

  // ---- 4) per-mode inverse FFT (real part), then scatter to out ----
#pragma unroll
  for (int k = 0; k < KM; ++k) {
    __syncthreads();
#pragma unroll
    for (int j = 0; j < 8; ++j) {
      const int f = tid * 8 + j;
      sRe[f] = ur_[k][j];
      sIm[f] = ui_[k][j];
    }
    __syncthreads();
    // stage 1: P[f0][t0] = sum_f1 N[f1][f0] * Wc[f1][t0]   (complex)
    {
      v8f pr = vzero8(), pi = vzero8();
      pr = mm64(pr, sRe, dftR, true,  1.f, mi, ni, lane);
      pr = mm64(pr, sIm, dftI, true, -1.f, mi, ni, lane);
      pi = mm64(pi, sRe, dftI, true,  1.f, mi, ni, lane);
      pi = mm64(pi, sIm, dftR, true,  1.f, mi, ni, lane);
      st64(tRe, pr, mi, ni, lane);
      st64(tIm, pi, mi, ni, lane);
    }
    __syncthreads();
    // inverse twiddle: Q = P .* e^{+2pi i t0 f0 / 4096}
#pragma unroll
    for (int j = 0; j < 8; ++j) {
      const int i = tid * 8 + j;
      const float pr = tRe[i], pi = tIm[i];
      const float wr = twR[i], wi = twI[i];
      tRe[i] = pr * wr - pi * wi;
      tIm[i] = pr * wi + pi * wr;
    }
    __syncthreads();
    // stage 2: s[t1][t0] = Re( sum_f0 Wc[f0][t1] * Q[f0][t0] )  (Wc symmetric)
    {
      v8f sr = vzero8();
      sr = mm64(sr, dftR, tRe, false,  1.f, mi, ni, lane);
      sr = mm64(sr, dftI, tIm, false, -1.f, mi, ni, lane);
      st64(sRe, sr, mi, ni, lane);
    }
    __syncthreads();
    // out[b, k, t, c] — 4B scatter @256B stride; write-combines in 192MB L2
    float* orow = out + (((size_t)b * KM + k) * T_LEN) * CCH + c;
    for (int i = tid; i < T_LEN; i += NTHR) orow[(size_t)i * CCH] = sRe[i];
  }
}

// ---------------------------------------------------------------------------
extern "C" void kernel_launch(void* const* d_in, const int* in_sizes, int n_in,
                              void* d_out, int out_size, void* d_ws, size_t ws_size,
                              hipStream_t stream) {
  const float* x     = (const float*)d_in[0];  // [32, 4096, 64]
  const float* alpha = (const float*)d_in[1];  // [8, 4]
  const float* tau   = (const float*)d_in[2];  // [8]
  const float* omega = (const float*)d_in[3];  // [4]
  float* tabs = (float*)d_ws;                  // 128 KB of tables
  float* out  = (float*)d_out;                 // [32, 4, 4096, 64]

  vmd_tables_kernel<<<16, 256, 0, stream>>>(tabs);

  const int rows = 32 * 64;                    // B * C = 2048 workgroups
  vmd_row_kernel<<<rows, NTHR, 0, stream>>>(x, alpha, tau, omega, tabs, out);
}